// GP_Layer_40991167873571
// MI455X (gfx1250) — compile-verified
//
#include <hip/hip_runtime.h>
#include <hip/hip_bf16.h>

// ---------------------------------------------------------------------------
// GA layer: y = mvlinear2( gp(h,h) ), h = mvlinear1(x);  out = a * y / mean||y||
// B=8 N=4096 Cin=64 Ch=128, 8 multivector components, grades [0,1,1,1,2,2,2,3]
// fp32 end-to-end via V_WMMA_F32_16X16X4_F32.
//
// Weight layout (prep kernel): Wt[g][kstep][o][klocal] with klocal in [0,4).
// A lane's B-fragment (k0, k0+1) is then ONE coalesced global_load_b64.
// ---------------------------------------------------------------------------

typedef __attribute__((ext_vector_type(2))) float v2f;
typedef __attribute__((ext_vector_type(8))) float v8f;
typedef __attribute__((ext_vector_type(4))) float f4;

__constant__ __device__ const int GRADES_C[8] = {0, 1, 1, 1, 2, 2, 2, 3};

// ---- compile-time Cayley table (matches the Python _build_cayley) ----------
struct CayT { signed char k[8][8]; signed char s[8][8]; };
constexpr int cpopc(int x) { int c = 0; while (x) { c += x & 1; x >>= 1; } return c; }
constexpr CayT buildCay() {
    CayT t{};
    const int ORD[8] = {0, 1, 2, 4, 3, 5, 6, 7};
    for (int ia = 0; ia < 8; ++ia)
        for (int ib = 0; ib < 8; ++ib) {
            int a = ORD[ia], b = ORD[ib];
            int q = a >> 1, tot = 0;
            while (q) { tot += cpopc(q & b); q >>= 1; }
            int res = a ^ b, k = 0;
            for (int j = 0; j < 8; ++j) if (ORD[j] == res) k = j;
            t.k[ia][ib] = (signed char)k;
            t.s[ia][ib] = (signed char)((tot % 2 == 0) ? 1 : -1);
        }
    return t;
}
constexpr CayT CAY = buildCay();

// ---------------------------------------------------------------------------
// Prep: repack weights so WMMA B-fragments are single b64 coalesced loads.
// W1t[g][kk][o][kl]: 4 x 16 x 128 x 4     (kk = K/4 step, kl = K%4)
// W2t[g][kk][o][kl]: 4 x 32 x  64 x 4
// ---------------------------------------------------------------------------
__global__ __launch_bounds__(256) void prep_w_kernel(const float* __restrict__ W1,
                                                     const float* __restrict__ W2,
                                                     float* __restrict__ W1t,
                                                     float* __restrict__ W2t) {
    int tid = blockIdx.x * 256 + threadIdx.x;
    if (tid < 32768) {
        int kl = tid & 3;
        int o  = (tid >> 2) & 127;
        int kk = (tid >> 9) & 15;
        int g  = tid >> 13;
        W1t[tid] = W1[(o * 64 + (kk * 4 + kl)) * 4 + g];
    } else if (tid < 65536) {
        int e  = tid - 32768;
        int kl = e & 3;
        int o  = (e >> 2) & 63;
        int kk = (e >> 8) & 31;
        int g  = e >> 13;
        W2t[e] = W2[(o * 128 + (kk * 4 + kl)) * 4 + g];
    }
}

// ---------------------------------------------------------------------------
// Main fused kernel: one block = 16 points, one wave = one component.
// ---------------------------------------------------------------------------
#define XS_STRIDE 66   // 64 + pad (even -> b64-aligned, bank-conflict-free)
#define HS_STRIDE 130  // 128 + pad

__global__ __launch_bounds__(256) void gp_layer_kernel(
    const float* __restrict__ x, const float* __restrict__ W1t,
    const float* __restrict__ b1, const float* __restrict__ W2t,
    const float* __restrict__ b2, const float* __restrict__ avec,
    float* __restrict__ out) {

    __shared__ __align__(16) float Xs[8 * 16 * XS_STRIDE];  // X tile, later reused for Y
    __shared__ __align__(16) float Hs[8 * 16 * HS_STRIDE];  // hidden tile (128 ch)
    __shared__ float red[16 * 16];

    const int t = threadIdx.x;
    const int wave = t >> 5;           // 0..7  == multivector component i
    const int lane = t & 31;
    const int mrow = lane & 15;        // A-matrix row / B-matrix column (per ISA layout)
    const int khalf = (lane >> 4) * 2; // lanes 16-31 carry K+2
    const int mbase = (lane >> 4) * 8; // C/D: lanes 16-31 hold rows r+8
    const long bn0 = (long)blockIdx.x * 16;

    // ---- Stage A: load x tile (16 x 64 x 8 fp32) into LDS as Xs[i][m][c] ----
    {
        const f4* src = (const f4*)(x + bn0 * 512);  // 512 floats per point
        #pragma unroll
        for (int r = 0; r < 8; ++r) {
            int idx4 = t + r * 256;        // float4 index, 2048 total
            int f0 = idx4 * 4;
            int m = f0 >> 9;
            int c = (f0 >> 3) & 63;
            int i0 = f0 & 7;               // 0 or 4 (never crosses c boundary)
            f4 v = __builtin_nontemporal_load(src + idx4);
            Xs[((i0 + 0) * 16 + m) * XS_STRIDE + c] = v.x;
            Xs[((i0 + 1) * 16 + m) * XS_STRIDE + c] = v.y;
            Xs[((i0 + 2) * 16 + m) * XS_STRIDE + c] = v.z;
            Xs[((i0 + 3) * 16 + m) * XS_STRIDE + c] = v.w;
        }
    }
    __syncthreads();

    // ---- Stage B: H[i] (16x128) = X[i] (16x64) @ W1f[i]^T ------------------
    // N-tile pairs: one A-fragment feeds two accumulators (ILP + fewer ds ops)
    {
        const int g = GRADES_C[wave];
        const float* arow = Xs + (wave * 16 + mrow) * XS_STRIDE;
        const float* wbase = W1t + (size_t)g * (16 * 128 * 4);
        for (int np = 0; np < 4; ++np) {
            const int nt = np * 2;
            v8f acc0 = {}, acc1 = {};
            const float* wb0 = wbase + (nt * 16 + mrow) * 4 + khalf;
            const float* wb1 = wbase + ((nt + 1) * 16 + mrow) * 4 + khalf;
            #pragma unroll
            for (int kk = 0; kk < 16; ++kk) {
                v2f af  = *(const v2f*)(arow + kk * 4 + khalf);
                v2f bf0 = *(const v2f*)(wb0 + kk * 128 * 4);
                v2f bf1 = *(const v2f*)(wb1 + kk * 128 * 4);
                acc0 = __builtin_amdgcn_wmma_f32_16x16x4_f32(
                    false, af, false, bf0, (short)0, acc0, false, false);
                acc1 = __builtin_amdgcn_wmma_f32_16x16x4_f32(
                    false, af, false, bf1, (short)0, acc1, false, false);
            }
            float ba0 = (wave == 0) ? b1[nt * 16 + mrow] : 0.0f;
            float ba1 = (wave == 0) ? b1[(nt + 1) * 16 + mrow] : 0.0f;
            #pragma unroll
            for (int r = 0; r < 8; ++r) {
                Hs[(wave * 16 + mbase + r) * HS_STRIDE + nt * 16 + mrow]       = acc0[r] + ba0;
                Hs[(wave * 16 + mbase + r) * HS_STRIDE + (nt + 1) * 16 + mrow] = acc1[r] + ba1;
            }
        }
    }
    __syncthreads();

    // ---- Stage C: geometric product h = gp(h, h), in place on Hs ----------
    {
        #pragma unroll
        for (int p = 0; p < 8; ++p) {
            int id = t + p * 256;       // 0..2047 = (m, ch) pairs
            int ch = id & 127;
            int m = id >> 7;
            float h[8], gout[8];
            #pragma unroll
            for (int i2 = 0; i2 < 8; ++i2) h[i2] = Hs[(i2 * 16 + m) * HS_STRIDE + ch];
            #pragma unroll
            for (int i2 = 0; i2 < 8; ++i2) gout[i2] = 0.0f;
            #pragma unroll
            for (int ia = 0; ia < 8; ++ia) {
                #pragma unroll
                for (int ib = 0; ib < 8; ++ib) {
                    if (CAY.s[ia][ib] > 0)
                        gout[(int)CAY.k[ia][ib]] = fmaf(h[ia], h[ib], gout[(int)CAY.k[ia][ib]]);
                    else
                        gout[(int)CAY.k[ia][ib]] = fmaf(-h[ia], h[ib], gout[(int)CAY.k[ia][ib]]);
                }
            }
            #pragma unroll
            for (int i2 = 0; i2 < 8; ++i2) Hs[(i2 * 16 + m) * HS_STRIDE + ch] = gout[i2];
        }
    }
    __syncthreads();

    // ---- Stage D: Y[i] (16x64) = H[i] (16x128) @ W2f[i]^T  (Y into Xs) -----
    {
        const int g = GRADES_C[wave];
        const float* arow = Hs + (wave * 16 + mrow) * HS_STRIDE;
        const float* wbase = W2t + (size_t)g * (32 * 64 * 4);
        for (int np = 0; np < 2; ++np) {
            const int nt = np * 2;
            v8f acc0 = {}, acc1 = {};
            const float* wb0 = wbase + (nt * 16 + mrow) * 4 + khalf;
            const float* wb1 = wbase + ((nt + 1) * 16 + mrow) * 4 + khalf;
            #pragma unroll
            for (int kk = 0; kk < 32; ++kk) {
                v2f af  = *(const v2f*)(arow + kk * 4 + khalf);
                v2f bf0 = *(const v2f*)(wb0 + kk * 64 * 4);
                v2f bf1 = *(const v2f*)(wb1 + kk * 64 * 4);
                acc0 = __builtin_amdgcn_wmma_f32_16x16x4_f32(
                    false, af, false, bf0, (short)0, acc0, false, false);
                acc1 = __builtin_amdgcn_wmma_f32_16x16x4_f32(
                    false, af, false, bf1, (short)0, acc1, false, false);
            }
            float ba0 = (wave == 0) ? b2[nt * 16 + mrow] : 0.0f;
            float ba1 = (wave == 0) ? b2[(nt + 1) * 16 + mrow] : 0.0f;
            #pragma unroll
            for (int r = 0; r < 8; ++r) {
                Xs[(wave * 16 + mbase + r) * XS_STRIDE + nt * 16 + mrow]       = acc0[r] + ba0;
                Xs[(wave * 16 + mbase + r) * XS_STRIDE + (nt + 1) * 16 + mrow] = acc1[r] + ba1;
            }
        }
    }
    __syncthreads();

    // ---- Stage E: per-point channel-mean norm, scale, store ----------------
    {
        int m = t >> 4;      // point in tile
        int cg = t & 15;     // channel group of 4
        float partial = 0.0f;
        #pragma unroll
        for (int j = 0; j < 4; ++j) {
            int c = cg * 4 + j;
            float ss = 0.0f;
            #pragma unroll
            for (int i2 = 0; i2 < 8; ++i2) {
                float v = Xs[(i2 * 16 + m) * XS_STRIDE + c];
                ss = fmaf(v, v, ss);
            }
            partial += sqrtf(ss);
        }
        red[m * 16 + cg] = partial;
    }
    __syncthreads();
    if ((t & 15) == 0) {
        int m = t >> 4;
        float s = 0.0f;
        #pragma unroll
        for (int j = 0; j < 16; ++j) s += red[m * 16 + j];
        red[m * 16] = 1.0f / (s * (1.0f / 64.0f) + 1e-6f);
    }
    __syncthreads();
    {
        float* dst = out + bn0 * 512;
        #pragma unroll
        for (int r = 0; r < 32; ++r) {
            int f = t + r * 256;       // 0..8191
            int m = f >> 9;
            int c = (f >> 3) & 63;
            int i2 = f & 7;
            float val = avec[c] * Xs[(i2 * 16 + m) * XS_STRIDE + c] * red[m * 16];
            __builtin_nontemporal_store(val, dst + f);
        }
    }
}

// ---------------------------------------------------------------------------
extern "C" void kernel_launch(void* const* d_in, const int* in_sizes, int n_in,
                              void* d_out, int out_size, void* d_ws, size_t ws_size,
                              hipStream_t stream) {
    const float* x  = (const float*)d_in[0];
    const float* W1 = (const float*)d_in[1];
    const float* b1 = (const float*)d_in[2];
    const float* W2 = (const float*)d_in[3];
    const float* b2 = (const float*)d_in[4];
    const float* av = (const float*)d_in[5];

    float* W1t = (float*)d_ws;           // 32768 floats
    float* W2t = W1t + 32768;            // 32768 floats (total 256 KB of d_ws)

    prep_w_kernel<<<256, 256, 0, stream>>>(W1, W2, W1t, W2t);

    // B*N / 16 = 32768 / 16 = 2048 tiles
    gp_layer_kernel<<<2048, 256, 0, stream>>>(x, W1t, b1, W2t, b2, av, (float*)d_out);
}